// LPTNTorch_38749194944737
// MI455X (gfx1250) — compile-verified
//
#include <hip/hip_runtime.h>
#include <math.h>

typedef float v4f __attribute__((ext_vector_type(4)));

// ---------------- derived-constant layout in d_ws (float[34]) ----------------
// 0..3   TS * inv_caps (softplus, TS folded in)
// 4..6   1/Rs_sy_sw, 1/Rs_sy_st, 1/Rs_sw_st
// 7      lin_slope = -sp(c7)
// 8      lin_bias  =  sp(c8)
// 9..11  m0,m1,m2
// 12..14 e0,e1,e2  = -log2(e)/b_i      (so exp(-sn/b) == exp2(sn*e_i))
// 15..17 a0,a1,a2
// 18..21 bp_magn(sp), bp_a, bp_b, bp_c (raw)
// 22     pdc = 1.5*Rdc                  (ploss_dc_ref = 3*Rdc*(cur/sqrt2)^2)
// 23..26 alpha_cu, ac1, ac2, beta_cu
// 27..30 k10, k11, k12*sqrt(.5)/256, k13*sqrt(.5)/256   (fold nc scaling)
// 31     k2c = sigmoid(c31)
// 32     alpha_fe = -sp(c32)
// 33     schlepp_f

// One coefficient per lane: the whole transform runs as ~1 vectorized softplus
// instead of 34 serialized scalar ones.
__global__ void lptn_prep(const float* __restrict__ c, float* __restrict__ cs) {
    const int i = blockIdx.x * blockDim.x + threadIdx.x;
    if (i >= 34) return;
    const float L2E = 1.44269504088896340736f;
    const float ncs = 0.70710678118654752440f / 256.0f;  // (1/sqrt2)/256
    const float TSf = 0.5f;

    const float x = c[i];
    const float s = log1pf(expf(x));  // softplus (vector VALU path)
    float r;
    if (i <= 3)       r = TSf * s;                        // TS * inv_caps
    else if (i <= 6)  r = 1.0f / s;                       // 1/Rs_*
    else if (i == 7)  r = -s;                             // lin_slope
    else if (i <= 11) r = s;                              // lin_bias, m0..m2
    else if (i <= 14) r = -L2E / s;                       // e0..e2
    else if (i <= 18) r = s;                              // a0..a2, bp_magn
    else if (i <= 21) r = x;                              // bp_a, bp_b, bp_c raw
    else if (i == 22) r = 1.5f * s;                       // pdc = 1.5*Rdc
    else if (i <= 26) r = s;                              // alpha_cu, ac1, ac2, beta_cu
    else if (i <= 28) r = x;                              // k10, k11 raw
    else if (i <= 30) r = x * ncs;                        // k12, k13 folded
    else if (i == 31) r = 1.0f / (1.0f + expf(-x));       // k2c sigmoid
    else if (i == 32) r = -s;                             // alpha_fe
    else              r = s;                              // schlepp_f
    cs[i] = r;
}

// ---------------- main elementwise kernel ----------------

struct Cs {
    float ic0, ic1, ic2, ic3;                       // TS*inv_cap
    float irs_sy_sw, irs_sy_st, irs_sw_st;
    float lin_slope, lin_bias;
    float m0, m1, m2, e0, e1, e2, a0, a1, a2;
    float bpm, bpa, bpb, bpc;
    float pdc, alpha_cu, ac1, ac2, beta_cu;
    float k10, k11, k12, k13, k2c, alpha_fe, schlepp_f;
};

__device__ __forceinline__ float fexp2(float x) { return __builtin_amdgcn_exp2f(x); }
__device__ __forceinline__ float flog2(float x) { return __builtin_amdgcn_logf(x); }
__device__ __forceinline__ float frcp(float x)  { return __builtin_amdgcn_rcpf(x); }

__device__ __forceinline__ v4f lptn_row(float amb, float cool, float speed,
                                        float cur, float ploss_meas, float schlepp,
                                        v4f h, const Cs& C) {
    const float th_pm = h.x, th_sy = h.y, th_st = h.z, th_sw = h.w;

    const float r_c_sy   = C.lin_bias * fmaf(C.lin_slope, cool - 20.0f, 1.0f);
    const float sn       = __builtin_fabsf(speed) * (1.0f / 6000.0f);
    const float r_st_pm  = fmaf(C.m0, fexp2(sn * C.e0), C.a0);
    const float r_sw_pm  = fmaf(C.m1, fexp2(sn * C.e1), C.a1);
    const float r_pm_amb = fmaf(C.m2, fexp2(sn * C.e2), C.a2);
    const float cn       = cool * 0.01f;
    const float r_pm_c   = fmaxf(fmaf(fmaf(C.bpc, cn, C.bpa), sn, fmaf(C.bpb, cn, C.bpm)), 1e-6f);

    const float r_ac_m1 = sn * fmaf(C.ac2, sn, C.ac1);                 // Rac/Rdc - 1
    const float lin     = fmaf(C.alpha_cu, th_sw - 70.0f, 1.0f);
    const float pdc     = C.pdc * cur * cur;                            // 3*Rdc*(cur/sqrt2)^2
    const float pcu_ref = pdc * (1.0f + r_ac_m1);
    const float xb      = fmaxf(__builtin_fabsf(lin), 1e-4f);
    const float denom   = fmaxf(fexp2(C.beta_cu * flog2(xb)), 1e-5f);
    const float ploss_sw = fmaf(pdc, lin, pcu_ref * r_ac_m1 * frcp(denom));

    const float ploss_fe = ploss_meas - C.schlepp_f * schlepp - pcu_ref;
    float k1 = fmaf(fmaf(C.k13, sn, C.k12), cur, fmaf(C.k11, sn, C.k10));
    k1 = fminf(fmaxf(k1, 0.0f), 1.0f);
    const float pf_k1 = k1 * ploss_fe;
    const float ploss_pm = (ploss_fe - pf_k1) * fmaf(C.alpha_fe, th_pm - 63.0f, 1.0f);
    const float ploss_sy = (C.k2c * pf_k1) * fmaf(C.alpha_fe, th_sy - 55.0f, 1.0f);
    const float ploss_st = ((1.0f - C.k2c) * pf_k1) * fmaf(C.alpha_fe, th_st - 65.0f, 1.0f);

    const float inv_st_pm = frcp(r_st_pm);
    const float inv_sw_pm = frcp(r_sw_pm);
    const float pm_d = (th_st - th_pm) * inv_st_pm + (th_sw - th_pm) * inv_sw_pm
                     + (amb - th_pm) * frcp(r_pm_amb) + (cool - th_pm) * frcp(r_pm_c);
    const float sy_d = (th_sw - th_sy) * C.irs_sy_sw + (th_st - th_sy) * C.irs_sy_st
                     + (cool - th_sy) * frcp(r_c_sy);
    const float st_d = (th_sy - th_st) * C.irs_sy_st + (th_sw - th_st) * C.irs_sw_st
                     + (th_pm - th_st) * inv_st_pm;
    const float sw_d = (th_sy - th_sw) * C.irs_sy_sw + (th_st - th_sw) * C.irs_sw_st
                     + (th_pm - th_sw) * inv_sw_pm;

    v4f o;
    o.x = fmaf(C.ic0, pm_d + ploss_pm, th_pm);   // TS already folded into ic*
    o.y = fmaf(C.ic1, sy_d + ploss_sy, th_sy);
    o.z = fmaf(C.ic2, st_d + ploss_st, th_st);
    o.w = fmaf(C.ic3, sw_d + ploss_sw, th_sw);
    return o;
}

__device__ __forceinline__ Cs load_cs(const float* __restrict__ cs) {
    Cs C;
    C.ic0 = cs[0];  C.ic1 = cs[1];  C.ic2 = cs[2];  C.ic3 = cs[3];
    C.irs_sy_sw = cs[4]; C.irs_sy_st = cs[5]; C.irs_sw_st = cs[6];
    C.lin_slope = cs[7]; C.lin_bias = cs[8];
    C.m0 = cs[9];  C.m1 = cs[10]; C.m2 = cs[11];
    C.e0 = cs[12]; C.e1 = cs[13]; C.e2 = cs[14];
    C.a0 = cs[15]; C.a1 = cs[16]; C.a2 = cs[17];
    C.bpm = cs[18]; C.bpa = cs[19]; C.bpb = cs[20]; C.bpc = cs[21];
    C.pdc = cs[22]; C.alpha_cu = cs[23]; C.ac1 = cs[24]; C.ac2 = cs[25];
    C.beta_cu = cs[26];
    C.k10 = cs[27]; C.k11 = cs[28]; C.k12 = cs[29]; C.k13 = cs[30];
    C.k2c = cs[31]; C.alpha_fe = cs[32]; C.schlepp_f = cs[33];
    return C;
}

__global__ __launch_bounds__(256) void lptn_main(
    const float* __restrict__ inp, const float* __restrict__ hidden,
    const float* __restrict__ cs, float* __restrict__ out, int nrows) {
    const Cs C = load_cs(cs);

    const long long t = (long long)blockIdx.x * blockDim.x + threadIdx.x;
    const long long row0 = t * 2;
    if (row0 >= nrows) return;

    if (row0 + 2 <= nrows) {
        // fast path: 2 rows, all traffic 16B-aligned B128, non-temporal (streamed)
        const v4f* ip = (const v4f*)(inp + row0 * 6);
        const v4f i0 = __builtin_nontemporal_load(ip + 0);
        const v4f i1 = __builtin_nontemporal_load(ip + 1);
        const v4f i2 = __builtin_nontemporal_load(ip + 2);
        const v4f* hp = (const v4f*)(hidden + row0 * 4);
        const v4f h0 = __builtin_nontemporal_load(hp + 0);
        const v4f h1 = __builtin_nontemporal_load(hp + 1);

        const v4f o0 = lptn_row(i0.x, i0.y, i0.z, i0.w, i1.x, i1.y, h0, C);
        const v4f o1 = lptn_row(i1.z, i1.w, i2.x, i2.y, i2.z, i2.w, h1, C);

        v4f* op = (v4f*)(out + row0 * 4);
        __builtin_nontemporal_store(o0, op + 0);
        __builtin_nontemporal_store(o1, op + 1);
    } else {
        // tail: single row, scalar loads
        const float* ir = inp + row0 * 6;
        const float* hr = hidden + row0 * 4;
        v4f h; h.x = hr[0]; h.y = hr[1]; h.z = hr[2]; h.w = hr[3];
        const v4f o = lptn_row(ir[0], ir[1], ir[2], ir[3], ir[4], ir[5], h, C);
        float* orow = out + row0 * 4;
        orow[0] = o.x; orow[1] = o.y; orow[2] = o.z; orow[3] = o.w;
    }
}

extern "C" void kernel_launch(void* const* d_in, const int* in_sizes, int n_in,
                              void* d_out, int out_size, void* d_ws, size_t ws_size,
                              hipStream_t stream) {
    const float* inp    = (const float*)d_in[0];   // [B,6]
    const float* hidden = (const float*)d_in[1];   // [B,4]
    const float* coeffs = (const float*)d_in[2];   // [34]
    float* out = (float*)d_out;                    // [B,4]
    float* cs  = (float*)d_ws;                     // 34 derived constants

    const int nrows = in_sizes[1] / 4;

    lptn_prep<<<1, 64, 0, stream>>>(coeffs, cs);   // 2 waves, 1 coeff/lane

    const long long nthreads = ((long long)nrows + 1) / 2;
    const int block = 256;
    const long long grid = (nthreads + block - 1) / block;
    lptn_main<<<(unsigned)grid, block, 0, stream>>>(inp, hidden, cs, out, nrows);
}